// GCNLayer_12893491822858
// MI455X (gfx1250) — compile-verified
//
#include <hip/hip_runtime.h>

typedef __attribute__((ext_vector_type(2))) float v2f;
typedef __attribute__((ext_vector_type(8))) float v8f;

#define N_NODES   50000
#define N_EDGES   800000
#define FEATS     100           // IN_FEATS == OUT_FEATS == 100
#define K_STEPS   25            // 100 / 4
#define M_TILES   3125          // 50000 / 16
#define N_TILES   7             // ceil(100/16) -> 112 padded
#define N_PAD     112
#define TILES     (M_TILES * N_TILES)

// ---------------- 1) zero the workspace (sum + deg) ----------------
__global__ void gcn_zero_kernel(float* __restrict__ p, int n) {
    int i = blockIdx.x * blockDim.x + threadIdx.x;
    if (i < n) p[i] = 0.0f;
}

// ---------------- 2) edge scatter: wave per edge, lane per feature ----------------
__global__ __launch_bounds__(256) void gcn_edge_scatter(
        const float* __restrict__ feat,
        const int*   __restrict__ src,
        const int*   __restrict__ dst,
        float* __restrict__ sum,
        float* __restrict__ deg) {
    int gtid = blockIdx.x * blockDim.x + threadIdx.x;
    int e    = gtid >> 5;          // one wave32 per edge
    int lane = gtid & 31;
    if (e >= N_EDGES) return;
    int s = src[e];
    int d = dst[e];
    const float* frow = feat + (long long)s * FEATS;
    float*       orow = sum  + (long long)d * FEATS;
    // 100 feats striped over 32 lanes: 4 coalesced rounds (last partial)
    for (int f = lane; f < FEATS; f += 32)
        atomicAdd(orow + f, frow[f]);
    if (lane == 0)
        atomicAdd(deg + d, 1.0f);
}

// ---------------- 3) fused mean + GEMM + bias via V_WMMA_F32_16X16X4_F32 ----------------
// out[m][n] = sum_k (sum[m][k] * invdeg[m]) * W[k][n] + bias[n]
__global__ __launch_bounds__(256) void gcn_wmma_gemm(
        const float* __restrict__ sum,
        const float* __restrict__ deg,
        const float* __restrict__ W,
        const float* __restrict__ bias,
        float* __restrict__ out) {
    // Stage weight in LDS, padded to 112 columns (zeros) so the nt==6 tile
    // can load without predication. 100*112*4 = 44.8 KB.
    __shared__ float ldsW[FEATS][N_PAD];
    for (int i = threadIdx.x; i < FEATS * N_PAD; i += blockDim.x) {
        int k = i / N_PAD;
        int n = i % N_PAD;
        ldsW[k][n] = (n < FEATS) ? W[k * FEATS + n] : 0.0f;
    }
    __syncthreads();

    int wave = threadIdx.x >> 5;
    int lane = threadIdx.x & 31;
    int tile = blockIdx.x * 8 + wave;      // 8 waves / block, 1 tile / wave
    if (tile >= TILES) return;             // wave-uniform: EXEC stays all-1s

    int mt   = tile / N_TILES;
    int nt   = tile % N_TILES;
    int half = lane >> 4;                  // 0: K pair {0,1}, 1: K pair {2,3}
    int lid  = lane & 15;

    int m = mt * 16 + lid;                 // A row held by this lane
    int n = nt * 16 + lid;                 // B/D column held by this lane

    // fold segment-mean into the A fragment (per-row scale, loop-invariant)
    float invd = 1.0f / fmaxf(deg[m], 1.0f);
    const float* arow = sum + (long long)m * FEATS;

    v8f c = {0.f, 0.f, 0.f, 0.f, 0.f, 0.f, 0.f, 0.f};

    for (int k0 = 0; k0 < FEATS; k0 += 4) {
        int k = k0 + half * 2;
        // A 16x4 f32 layout: lanes 0-15 -> K=k0,k0+1 ; lanes 16-31 -> K=k0+2,k0+3
        v2f a = *(const v2f*)(arow + k);   // m*100 + even k -> 8B aligned
        a.x *= invd;
        a.y *= invd;
        // B 4x16 f32 layout mirrors A's K striping; row striped across lanes
        v2f b;
        b.x = ldsW[k][n];
        b.y = ldsW[k + 1][n];
        c = __builtin_amdgcn_wmma_f32_16x16x4_f32(
                /*neg_a=*/false, a, /*neg_b=*/false, b,
                /*c_mod=*/(short)0, c, /*reuse_a=*/false, /*reuse_b=*/false);
    }

    // D layout: VGPR i -> row m = mt*16 + i + 8*half, col n (lanes 0-15 / 16-31)
    if (n < FEATS) {
        float bn = bias[n];
        #pragma unroll
        for (int i = 0; i < 8; ++i) {
            int row = mt * 16 + i + half * 8;
            out[(long long)row * FEATS + n] = c[i] + bn;
        }
    }
}

extern "C" void kernel_launch(void* const* d_in, const int* in_sizes, int n_in,
                              void* d_out, int out_size, void* d_ws, size_t ws_size,
                              hipStream_t stream) {
    const float* feat = (const float*)d_in[0];
    const int*   src  = (const int*)  d_in[1];
    const int*   dst  = (const int*)  d_in[2];
    const float* W    = (const float*)d_in[3];
    const float* bias = (const float*)d_in[4];
    float*       out  = (float*)d_out;

    float* sum = (float*)d_ws;                       // [50000 x 100]
    float* deg = sum + (size_t)N_NODES * FEATS;      // [50000]

    int ztot = N_NODES * FEATS + N_NODES;            // 5,050,000 floats
    gcn_zero_kernel<<<(ztot + 255) / 256, 256, 0, stream>>>(sum, ztot);

    int edge_threads = N_EDGES * 32;                 // wave per edge
    gcn_edge_scatter<<<(edge_threads + 255) / 256, 256, 0, stream>>>(
        feat, src, dst, sum, deg);

    gcn_wmma_gemm<<<(TILES + 7) / 8, 256, 0, stream>>>(
        sum, deg, W, bias, out);
}